// torch_specular_renderer_59416577573398
// MI455X (gfx1250) — compile-verified
//
#include <hip/hip_runtime.h>
#include <math.h>

typedef float v2f __attribute__((ext_vector_type(2)));
typedef float v8f __attribute__((ext_vector_type(8)));

#define EHh 16
#define EWw 32
#define MDIRS 512                    // EH*EW light directions
#define HWPIX (512 * 512)            // pixels per image
#define NB 2                         // batch
#define NTILES ((NB * HWPIX) / 16)   // 32768 pixel tiles of 16
#define DIRTILES (MDIRS / 16)        // 32 dir tiles of 16
#define WAVES (NTILES / 2)           // 2 pixel tiles per wave

// ws float-offset layout:
//  [0]      : unsigned gmax bits
//  [4)      : hA   [32 dir-tiles][2 vgprs][32 lanes]          (8 KB)
//  [2052)   : hdrA [NB][32 j][4 t][2 vgprs][32 lanes]         (64 KB)
#define OFF_HA 4
#define OFF_HDRA (4 + DIRTILES * 2 * 32)

// ---------------------------------------------------------------------------
// Kernel 1: constants (one block, 512 threads = one light dir each).
// ---------------------------------------------------------------------------
__global__ __launch_bounds__(512) void init_consts(const float* __restrict__ env,
                                                   float* __restrict__ ws) {
  const int m = threadIdx.x;
  if (m == 0) ((unsigned*)ws)[0] = 0u;

  const int eh = m / EWw, ew = m % EWw;
  const float phi = (float)eh / (float)EHh * 3.14159265358979323846f;
  const float theta = (float)ew / (float)EWw * 6.28318530717958647692f;
  const float sp = sinf(phi), cp = cosf(phi);
  const float st = sinf(theta), ct = cosf(theta);
  // h = normalize(v + l), v=(0,0,1), l=(st*sp, cp, -ct*sp)
  float hx = st * sp, hy = cp, hz = 1.0f - ct * sp;
  const float hinv = __builtin_amdgcn_rsqf(hx * hx + hy * hy + hz * hz);
  hx *= hinv; hy *= hinv; hz *= hinv;

  const int j = m >> 4, r = m & 15;

  // First-GEMM A operand (16 dirs x K=4), 32-bit 16x4 A layout:
  //  lanes 0-15: VGPR0=K0(h.x), VGPR1=K1(h.y); lanes 16-31: VGPR0=K2(h.z), VGPR1=K3(0)
  float* hA = ws + OFF_HA;
  hA[(j * 2 + 0) * 32 + r]      = hx;
  hA[(j * 2 + 0) * 32 + 16 + r] = hz;
  hA[(j * 2 + 1) * 32 + r]      = hy;
  hA[(j * 2 + 1) * 32 + 16 + r] = 0.0f;

  // Second-GEMM A operand: hdr[ch][dir], K-rows permuted so first GEMM's C
  // VGPR pair (2t,2t+1) is directly the B operand of K-chunk t:
  //  Klocal {0,1,2,3} <-> dir rows {2t, 2t+1, 2t+8, 2t+9}.
  const int t = (r >> 1) & 3;
  const int vg = r & 1;
  const int up = (r & 8) ? 16 : 0;
  float* hdrA = ws + OFF_HDRA;
  for (int b = 0; b < NB; ++b) {
    const float* e = env + ((b * EHh + eh) * EWw + ew) * 3;
    float* dst = hdrA + ((((b * DIRTILES + j) * 4 + t) * 2 + vg) * 32 + up);
#pragma unroll
    for (int ch = 0; ch < 16; ++ch)
      dst[ch] = (ch < 3) ? e[ch] * sp : 0.0f;
  }
}

// B operand of first GEMM: 16 pixels' normals (K=4 x N=16 px), 32-bit B layout:
//  VGPR0 = {K0 row | K2 row}, VGPR1 = {K1 row | K3(0) row}
__device__ inline v2f load_Bn(const float* __restrict__ normal, int tile, int lane) {
  const int p = tile * 16 + (lane & 15);
  const float r0 = normal[p * 3 + 0];
  const float r1 = normal[p * 3 + 1];
  const float r2 = normal[p * 3 + 2];
  const float n0 = (r2 - 0.5f) * 2.0f;   // channel flip + [0,1]->[-1,1]
  const float n1 = (r1 - 0.5f) * 2.0f;
  const float n2 = (r0 - 0.5f) * 2.0f;
  const float nrm = sqrtf(n0 * n0 + n1 * n1 + n2 * n2);
  const float inv = __builtin_amdgcn_rcpf(fmaxf(nrm, 1e-12f));
  v2f b;
  b.x = (lane < 16) ? n0 * inv : n2 * inv;
  b.y = (lane < 16) ? n1 * inv : 0.0f;
  return b;
}

// ---------------------------------------------------------------------------
// Kernel 2: global max of relu(n.h); 2 pixel tiles/wave, A kept in registers,
// reuse-A hint across the tile pair, LDS block-level max before global atomic.
// ---------------------------------------------------------------------------
__global__ __launch_bounds__(256) void pass_max(const float* __restrict__ normal,
                                                const float* __restrict__ ws,
                                                unsigned* __restrict__ gmax) {
  __shared__ unsigned smax;
  if (threadIdx.x == 0) smax = 0u;

  const int lane = threadIdx.x & 31;
  const int wave = blockIdx.x * 8 + (threadIdx.x >> 5);
  const int tile0 = wave * 2;
  const float* hA = ws + OFF_HA;

  v2f aAll[DIRTILES];
#pragma unroll
  for (int j = 0; j < DIRTILES; ++j) {
    aAll[j].x = hA[(j * 2 + 0) * 32 + lane];
    aAll[j].y = hA[(j * 2 + 1) * 32 + lane];
  }
  const v2f bn0 = load_Bn(normal, tile0 + 0, lane);
  const v2f bn1 = load_Bn(normal, tile0 + 1, lane);
  __syncthreads();

  float mx = 0.0f;  // relu folded in
#pragma unroll
  for (int j = 0; j < DIRTILES; ++j) {
    v8f c0 = {}, c1 = {};
    c0 = __builtin_amdgcn_wmma_f32_16x16x4_f32(false, aAll[j], false, bn0,
                                               (short)0, c0, true, false);  // reuse A
    c1 = __builtin_amdgcn_wmma_f32_16x16x4_f32(false, aAll[j], false, bn1,
                                               (short)0, c1, false, false);
#pragma unroll
    for (int v = 0; v < 8; ++v) mx = fmaxf(mx, fmaxf(c0[v], c1[v]));
  }
#pragma unroll
  for (int off = 16; off >= 1; off >>= 1)
    mx = fmaxf(mx, __shfl_xor(mx, off, 32));
  if (lane == 0) atomicMax(&smax, __float_as_uint(mx));  // mx>=0: bit max == float max
  __syncthreads();
  if (threadIdx.x == 0) atomicMax(gmax, smax);
}

// ---------------------------------------------------------------------------
// Kernel 3: scores -> (s/max)^64 -> chained WMMA vs hdr -> (B,3,H,W)/60.
// 2 pixel tiles/wave; every A operand feeds two back-to-back WMMAs (reuse-A).
// ---------------------------------------------------------------------------
__global__ __launch_bounds__(256) void pass_render(const float* __restrict__ normal,
                                                   const float* __restrict__ ws,
                                                   const unsigned* __restrict__ gmax,
                                                   float* __restrict__ out) {
  const int lane = threadIdx.x & 31;
  const int wave = blockIdx.x * 8 + (threadIdx.x >> 5);
  const int tile0 = wave * 2;                 // tile pairs never straddle a batch
  const int batch = (tile0 * 16) / HWPIX;
  const float* hA = ws + OFF_HA;
  const float* hdrA = ws + OFF_HDRA + (size_t)batch * DIRTILES * 4 * 2 * 32;

  const float invmax = __builtin_amdgcn_rcpf(__uint_as_float(*gmax));
  const v2f bnA = load_Bn(normal, tile0 + 0, lane);
  const v2f bnB = load_Bn(normal, tile0 + 1, lane);

  v8f d0a = {}, d1a = {}, d0b = {}, d1b = {};   // 16ch x 16px accumulators
#pragma unroll 2
  for (int j = 0; j < DIRTILES; ++j) {
    v2f a1;
    a1.x = hA[(j * 2 + 0) * 32 + lane];
    a1.y = hA[(j * 2 + 1) * 32 + lane];
    v8f cA = {}, cB = {};
    cA = __builtin_amdgcn_wmma_f32_16x16x4_f32(false, a1, false, bnA,
                                               (short)0, cA, true, false);   // reuse A
    cB = __builtin_amdgcn_wmma_f32_16x16x4_f32(false, a1, false, bnB,
                                               (short)0, cB, false, false);
    // relu, normalize by global max, ^64 via 6 squarings (in place)
#pragma unroll
    for (int v = 0; v < 8; ++v) {
      float s = fmaxf(cA[v], 0.0f) * invmax;
      s = s * s; s = s * s; s = s * s;
      s = s * s; s = s * s; s = s * s;
      cA[v] = s;
      float u = fmaxf(cB[v], 0.0f) * invmax;
      u = u * u; u = u * u; u = u * u;
      u = u * u; u = u * u; u = u * u;
      cB[v] = u;
    }
    // four K=4 chunks: C-output VGPR pair (2t,2t+1) is directly the B operand
#pragma unroll
    for (int t = 0; t < 4; ++t) {
      v2f a2;
      const float* src = hdrA + ((j * 4 + t) * 2) * 32;
      a2.x = src[lane];
      a2.y = src[32 + lane];
      v2f bpA, bpB;
      bpA.x = cA[2 * t]; bpA.y = cA[2 * t + 1];
      bpB.x = cB[2 * t]; bpB.y = cB[2 * t + 1];
      if (t & 1) {
        d1a = __builtin_amdgcn_wmma_f32_16x16x4_f32(false, a2, false, bpA,
                                                    (short)0, d1a, true, false);
        d1b = __builtin_amdgcn_wmma_f32_16x16x4_f32(false, a2, false, bpB,
                                                    (short)0, d1b, false, false);
      } else {
        d0a = __builtin_amdgcn_wmma_f32_16x16x4_f32(false, a2, false, bpA,
                                                    (short)0, d0a, true, false);
        d0b = __builtin_amdgcn_wmma_f32_16x16x4_f32(false, a2, false, bpB,
                                                    (short)0, d0b, false, false);
      }
    }
  }

  // D VGPR v, lanes 0-15 = channel v of 16 pixels -> coalesced stores
  if (lane < 16) {
    const int pix = ((tile0 * 16) & (HWPIX - 1)) + lane;
    float* o = out + (size_t)batch * 3 * HWPIX + pix;
    o[0 * HWPIX]      = (d0a[0] + d1a[0]) * (1.0f / 60.0f);
    o[1 * HWPIX]      = (d0a[1] + d1a[1]) * (1.0f / 60.0f);
    o[2 * HWPIX]      = (d0a[2] + d1a[2]) * (1.0f / 60.0f);
    o[0 * HWPIX + 16] = (d0b[0] + d1b[0]) * (1.0f / 60.0f);
    o[1 * HWPIX + 16] = (d0b[1] + d1b[1]) * (1.0f / 60.0f);
    o[2 * HWPIX + 16] = (d0b[2] + d1b[2]) * (1.0f / 60.0f);
  }
}

extern "C" void kernel_launch(void* const* d_in, const int* in_sizes, int n_in,
                              void* d_out, int out_size, void* d_ws, size_t ws_size,
                              hipStream_t stream) {
  const float* env = (const float*)d_in[0];     // (B,16,32,3) f32
  const float* normal = (const float*)d_in[1];  // (B,512,512,3) f32
  float* out = (float*)d_out;                   // (B,3,512,512) f32
  float* ws = (float*)d_ws;
  unsigned* gmax = (unsigned*)d_ws;

  init_consts<<<1, 512, 0, stream>>>(env, ws);
  pass_max<<<WAVES / 8, 256, 0, stream>>>(normal, ws, gmax);
  pass_render<<<WAVES / 8, 256, 0, stream>>>(normal, ws, gmax, out);
}